// ShiftModule_25606595018769
// MI455X (gfx1250) — compile-verified
//
#include <hip/hip_runtime.h>
#include <stdint.h>

// Neural-VM shift module for MI455X (gfx1250).
// Memory-bound (1.07 GB traffic -> ~46us at 23.3 TB/s). Bulk copy runs on the
// CDNA5 async global<->LDS DMA path (ASYNCcnt), fixup applied in LDS.

#define THREADS        256
#define ROWS_PER_BLOCK 8      // one wave32 per row
#define DCOL           512
#define TILE_FLOATS    (ROWS_PER_BLOCK * DCOL)          // 4096 floats = 16 KiB
#define VEC4_PER_TILE  (TILE_FLOATS / 4)                // 1024 float4
#define ITERS          (VEC4_PER_TILE / THREADS)        // 4 b128 ops per lane

__global__ __launch_bounds__(THREADS)
void shift_vm_kernel(const float* __restrict__ x, float* __restrict__ out, int nrows) {
    __shared__ float tile[TILE_FLOATS];

    const int tid  = threadIdx.x;
    const int lane = tid & 31;
    const int wave = tid >> 5;

    const long long row0 = (long long)blockIdx.x * ROWS_PER_BLOCK;
    if (row0 >= nrows) return;

    const float* gsrc = x   + row0 * DCOL;
    float*       gdst = out + row0 * DCOL;

    // Low 32 bits of a flat LDS-aperture address == LDS byte offset (ISA 10.2).
    const unsigned lds0 = (unsigned)(uintptr_t)(&tile[0]);

    // ---- Stage 1: async DMA tile global -> LDS (16B per lane per iter) ----
    #pragma unroll
    for (int k = 0; k < ITERS; ++k) {
        unsigned f    = (unsigned)tid + (unsigned)k * THREADS;  // float4 idx
        unsigned goff = f * 16u;
        unsigned loff = lds0 + f * 16u;
        asm volatile("global_load_async_to_lds_b128 %0, %1, %2"
                     :
                     : "v"(loff), "v"(goff), "s"(gsrc)
                     : "memory");
    }
    asm volatile("s_wait_asynccnt 0" ::: "memory");
    __syncthreads();

    // ---- Stage 2: per-row decode + in-LDS scatter; one wave per row ----
    {
        const int rb = wave * DCOL;
        const int li = lane & 15;

        // Combined argmax: lanes 0-15 reduce ALU_LO (cols 16..31),
        // lanes 16-31 reduce ALU_HI (cols 32..47). xor<=8 stays in-group.
        float v = tile[rb + ((lane < 16) ? (16 + li) : (32 + li))];
        int   i = li;
        #pragma unroll
        for (int off = 8; off >= 1; off >>= 1) {
            float ov = __shfl_xor(v, off, 32);
            int   oi = __shfl_xor(i, off, 32);
            if (ov > v || (ov == v && oi < i)) { v = ov; i = oi; }
        }
        const int a_lo = __shfl(i, 0, 32);
        const int a_hi = __shfl(i, 16, 32);

        // Shift amount field (cols 48..63); both halves compute it redundantly.
        float sv = tile[rb + 48 + li];
        int   si = li;
        #pragma unroll
        for (int off = 8; off >= 1; off >>= 1) {
            float ov = __shfl_xor(sv, off, 32);
            int   oi = __shfl_xor(si, off, 32);
            if (ov > sv || (ov == sv && oi < si)) { sv = ov; si = oi; }
        }
        int sh = __shfl(si, 0, 32);
        sh = sh > 7 ? 7 : sh;

        const float x0 = tile[rb + 0];
        const float x1 = tile[rb + 1];
        const float x2 = tile[rb + 2];
        const bool act_shl = (x0 > 0.5f) && (x2 > 0.5f);
        const bool act_shr = (x1 > 0.5f) && (x2 > 0.5f);

        // Exact integer semantics of the fp32 magic-floor shift.
        const int a   = a_lo + (a_hi << 4);               // 0..255
        const int res = act_shl ? ((a << sh) & 255) : (a >> sh);

        if (lane == 0 && (act_shl || act_shr)) {
            tile[rb + 64 + (res & 15)]        += 1.0f;    // OUTPUT_LO one-hot
            tile[rb + 80 + ((res >> 4) & 15)] += 1.0f;    // OUTPUT_HI one-hot
        }
    }
    asm volatile("s_wait_dscnt 0" ::: "memory");
    __syncthreads();

    // ---- Stage 3: async DMA tile LDS -> global ----
    #pragma unroll
    for (int k = 0; k < ITERS; ++k) {
        unsigned f    = (unsigned)tid + (unsigned)k * THREADS;
        unsigned goff = f * 16u;
        unsigned loff = lds0 + f * 16u;
        asm volatile("global_store_async_from_lds_b128 %0, %1, %2"
                     :
                     : "v"(goff), "v"(loff), "s"(gdst)
                     : "memory");
    }
    asm volatile("s_wait_asynccnt 0" ::: "memory");
}

extern "C" void kernel_launch(void* const* d_in, const int* in_sizes, int n_in,
                              void* d_out, int out_size, void* d_ws, size_t ws_size,
                              hipStream_t stream) {
    (void)n_in; (void)out_size; (void)d_ws; (void)ws_size;
    const float* x   = (const float*)d_in[0];
    float*       out = (float*)d_out;

    const int rows   = in_sizes[0] / DCOL;              // 262144
    const int blocks = rows / ROWS_PER_BLOCK;           // 32768
    shift_vm_kernel<<<blocks, THREADS, 0, stream>>>(x, out, rows);
}